// GridGenerator_Plus_73392401154541
// MI455X (gfx1250) — compile-verified
//
#include <hip/hip_runtime.h>
#include <hip/hip_bf16.h>

typedef float v2f __attribute__((ext_vector_type(2)));
typedef float v8f __attribute__((ext_vector_type(8)));

#define Bn 256
#define Ln 1024
#define Dn 64
#define Hn 4
#define Nn 64
#define NP 3200   // RH*RW = 32*100

// D = A(16x4 f32) * B(4x16 f32) + C(16x16 f32)
__device__ __forceinline__ v8f wmma_f32(v2f a, v2f b, v8f c) {
    return __builtin_amdgcn_wmma_f32_16x16x4_f32(
        false, a, false, b, (short)0, c, false, false);
}

// 16x64 row tile of A (row-major, stride 64) times 64x64 W (row-major):
// acc[ct] accumulates the 16x16 tile for output columns [16*ct, 16*ct+16).
__device__ __forceinline__ void gemm64(const float* A, const float* W,
                                       v8f acc[4], int lo, int hi) {
#pragma unroll
    for (int ks = 0; ks < 16; ++ks) {
        const int k0 = ks * 4;
        v2f a;
        const float* ap = A + (size_t)lo * 64 + k0 + 2 * hi;
        a.x = ap[0];
        a.y = ap[1];
#pragma unroll
        for (int ct = 0; ct < 4; ++ct) {
            v2f b;
            const float* bp = W + (size_t)(k0 + 2 * hi) * 64 + ct * 16 + lo;
            b.x = bp[0];
            b.y = bp[64];
            acc[ct] = wmma_f32(a, b, acc[ct]);
        }
    }
}

// ---------------- K1: fiducial query path (batch independent, tiny) --------
__global__ void __launch_bounds__(64) k_query(const float* __restrict__ W_emb,
                                              const float* __restrict__ b_emb,
                                              const float* __restrict__ Wq,
                                              const float* __restrict__ bq,
                                              float* __restrict__ q_out,
                                              float* __restrict__ qp_out) {
    __shared__ float qs[64 * 64];
    const int t = threadIdx.x;                 // fiducial row 0..63
    const float cx = -1.f + 2.f * (float)(t >> 2) / 15.f;  // linspace(-1,1,16)
    const float cy = -1.f + 2.f * (float)(t & 3) / 3.f;    // linspace(-1,1,4)
    for (int d = 0; d < 64; ++d) {
        float v = cx * W_emb[d] + cy * W_emb[64 + d] + b_emb[d];
        qs[t * 64 + d] = v;
        q_out[t * 64 + d] = v;
    }
    __syncthreads();
    for (int d = 0; d < 64; ++d) {
        float s = bq[d];
        for (int k = 0; k < 64; ++k) s += qs[t * 64 + k] * Wq[k * 64 + d];
        qp_out[t * 64 + d] = s;
    }
}

// ---------------- K2: fused kv = X@W_in, kp = kv@Wk, vp = kv@Wv ------------
__global__ void __launch_bounds__(128) k_kv(const float* __restrict__ X,
                                            const float* __restrict__ W_in,
                                            const float* __restrict__ b_in,
                                            const float* __restrict__ Wk,
                                            const float* __restrict__ bk,
                                            const float* __restrict__ Wv,
                                            const float* __restrict__ bv,
                                            float* __restrict__ kp_out,
                                            float* __restrict__ vp_out) {
    __shared__ float lds_kv[4][16 * 64];
    const int lane = threadIdx.x & 31;
    const int wave = threadIdx.x >> 5;
    const int lo = lane & 15, hi = lane >> 4;
    const size_t row0 = ((size_t)blockIdx.x * 4 + wave) * 16;  // 16-row tile
    const float* A = X + row0 * 64;
    __builtin_prefetch(A + 16 * 64, 0, 1);  // global_prefetch_b8 (next tile)

    v8f acc[4] = {};
    gemm64(A, W_in, acc, lo, hi);

    float* kv = lds_kv[wave];
#pragma unroll
    for (int ct = 0; ct < 4; ++ct) {
        const float bb = b_in[ct * 16 + lo];
#pragma unroll
        for (int j = 0; j < 8; ++j)
            kv[(j + 8 * hi) * 64 + ct * 16 + lo] = acc[ct][j] + bb;
    }

    v8f ak[4] = {};
    gemm64(kv, Wk, ak, lo, hi);
#pragma unroll
    for (int ct = 0; ct < 4; ++ct) {
        const float bb = bk[ct * 16 + lo];
#pragma unroll
        for (int j = 0; j < 8; ++j)
            kp_out[(row0 + j + 8 * hi) * 64 + ct * 16 + lo] = ak[ct][j] + bb;
    }

    v8f av[4] = {};
    gemm64(kv, Wv, av, lo, hi);
#pragma unroll
    for (int ct = 0; ct < 4; ++ct) {
        const float bb = bv[ct * 16 + lo];
#pragma unroll
        for (int j = 0; j < 8; ++j)
            vp_out[(row0 + j + 8 * hi) * 64 + ct * 16 + lo] = av[ct][j] + bb;
    }
}

// ---------------- K3: flash cross-attention --------------------------------
// grid = B*H blocks; wave w handles query rows [16w,16w+16) of that (b,h).
__global__ void __launch_bounds__(128) k_attn(const float* __restrict__ qp,
                                              const float* __restrict__ kp,
                                              const float* __restrict__ vp,
                                              float* __restrict__ attnO) {
    __shared__ float ldsP[4][16 * 16];
    const int lane = threadIdx.x & 31;
    const int wave = threadIdx.x >> 5;
    const int lo = lane & 15, hi = lane >> 4;
    const int b = blockIdx.x >> 2, h = blockIdx.x & 3;
    const int q0 = wave * 16;

    v2f aq[4];
#pragma unroll
    for (int ss = 0; ss < 4; ++ss) {
        const float* p = qp + (q0 + lo) * 64 + h * 16 + ss * 4 + 2 * hi;
        aq[ss].x = p[0];
        aq[ss].y = p[1];
    }
    const float* kpb = kp + (size_t)b * Ln * 64;
    const float* vpb = vp + (size_t)b * Ln * 64;

    v8f o = {};
    float m[8], l[8];
#pragma unroll
    for (int j = 0; j < 8; ++j) { m[j] = -3.0e38f; l[j] = 0.f; }
    const float scale = 0.25f;  // 1/sqrt(DK=16)
    float* P = ldsP[wave];

    for (int c = 0; c < Ln / 16; ++c) {
        const int l0 = c * 16;
        v8f s = {};
#pragma unroll
        for (int ss = 0; ss < 4; ++ss) {
            v2f bf;
            const float* p = kpb + (size_t)(l0 + lo) * 64 + h * 16 + ss * 4 + 2 * hi;
            bf.x = p[0];
            bf.y = p[1];
            s = wmma_f32(aq[ss], bf, s);
        }
        // online softmax per row (rows of this lane: M = j + 8*hi, cols N = lo)
#pragma unroll
        for (int j = 0; j < 8; ++j) {
            float sj = s[j] * scale;
            float mj = sj;
            mj = fmaxf(mj, __shfl_xor(mj, 8, 16));
            mj = fmaxf(mj, __shfl_xor(mj, 4, 16));
            mj = fmaxf(mj, __shfl_xor(mj, 2, 16));
            mj = fmaxf(mj, __shfl_xor(mj, 1, 16));
            const float mn = fmaxf(m[j], mj);
            const float f = __expf(m[j] - mn);
            float pj = __expf(sj - mn);
            float rs = pj;
            rs += __shfl_xor(rs, 8, 16);
            rs += __shfl_xor(rs, 4, 16);
            rs += __shfl_xor(rs, 2, 16);
            rs += __shfl_xor(rs, 1, 16);
            l[j] = l[j] * f + rs;
            m[j] = mn;
            o[j] *= f;
            P[(j + 8 * hi) * 16 + lo] = pj;  // stage P tile for A-layout reload
        }
        // O += P @ V_chunk  (A from LDS, B = vp rows l0..l0+15, cols = head dk)
#pragma unroll
        for (int ss = 0; ss < 4; ++ss) {
            v2f a;
            a.x = P[lo * 16 + ss * 4 + 2 * hi];
            a.y = P[lo * 16 + ss * 4 + 2 * hi + 1];
            v2f bf;
            const float* p = vpb + (size_t)(l0 + ss * 4 + 2 * hi) * 64 + h * 16 + lo;
            bf.x = p[0];
            bf.y = p[64];
            o = wmma_f32(a, bf, o);
        }
    }
    float* Ob = attnO + (size_t)b * 64 * 64;
#pragma unroll
    for (int j = 0; j < 8; ++j)
        Ob[(q0 + j + 8 * hi) * 64 + h * 16 + lo] = o[j] / l[j];
}

// ---------------- K4: Wo + residual + LN1 + FFN + LN2 + W_down -------------
__device__ __forceinline__ void lnorm(float* X, const float* g, const float* b,
                                      int tid) {
    if (tid < 64) {
        float mu = 0.f;
        for (int k = 0; k < 64; ++k) mu += X[tid * 64 + k];
        mu *= (1.f / 64.f);
        float var = 0.f;
        for (int k = 0; k < 64; ++k) {
            float d = X[tid * 64 + k] - mu;
            var += d * d;
        }
        var *= (1.f / 64.f);
        const float rs = rsqrtf(var + 1e-5f);
        for (int k = 0; k < 64; ++k)
            X[tid * 64 + k] = (X[tid * 64 + k] - mu) * rs * g[k] + b[k];
    }
}

__global__ void __launch_bounds__(128) k_post(
    const float* __restrict__ attnO, const float* __restrict__ q,
    const float* __restrict__ Wo, const float* __restrict__ bo,
    const float* __restrict__ ln1g, const float* __restrict__ ln1b,
    const float* __restrict__ W1, const float* __restrict__ b1,
    const float* __restrict__ W2, const float* __restrict__ b2,
    const float* __restrict__ ln2g, const float* __restrict__ ln2b,
    const float* __restrict__ Wd, const float* __restrict__ bd,
    float* __restrict__ Cout) {
    __shared__ float X[64 * 64];
    __shared__ float Hb[64 * 64];
    const int tid = threadIdx.x;
    const int lane = tid & 31, wave = tid >> 5;
    const int lo = lane & 15, hi = lane >> 4;
    const int b = blockIdx.x;
    const int r0 = wave * 16;

    v8f acc[4] = {};
    gemm64(attnO + (size_t)b * 4096 + r0 * 64, Wo, acc, lo, hi);
#pragma unroll
    for (int ct = 0; ct < 4; ++ct) {
        const int col = ct * 16 + lo;
#pragma unroll
        for (int j = 0; j < 8; ++j) {
            const int row = r0 + j + 8 * hi;
            X[row * 64 + col] = acc[ct][j] + bo[col] + q[row * 64 + col];
        }
    }
    __syncthreads();
    lnorm(X, ln1g, ln1b, tid);
    __syncthreads();

    v8f a1[4] = {};
    gemm64(X + r0 * 64, W1, a1, lo, hi);
#pragma unroll
    for (int ct = 0; ct < 4; ++ct) {
        const int col = ct * 16 + lo;
#pragma unroll
        for (int j = 0; j < 8; ++j)
            Hb[(r0 + j + 8 * hi) * 64 + col] = fmaxf(a1[ct][j] + b1[col], 0.f);
    }
    v8f a2[4] = {};
    gemm64(Hb + r0 * 64, W2, a2, lo, hi);
#pragma unroll
    for (int ct = 0; ct < 4; ++ct) {
        const int col = ct * 16 + lo;
#pragma unroll
        for (int j = 0; j < 8; ++j) {
            const int row = r0 + j + 8 * hi;
            X[row * 64 + col] = X[row * 64 + col] + a2[ct][j] + b2[col];
        }
    }
    __syncthreads();
    lnorm(X, ln2g, ln2b, tid);
    __syncthreads();

    if (tid < 64) {
        float s0 = bd[0], s1 = bd[1];
        for (int k = 0; k < 64; ++k) {
            const float v = X[tid * 64 + k];
            s0 += v * Wd[k * 2];
            s1 += v * Wd[k * 2 + 1];
        }
        Cout[((size_t)b * 64 + tid) * 2] = s0;
        Cout[((size_t)b * 64 + tid) * 2 + 1] = s1;
    }
}

// ---------------- K5a: cross-batch pairwise RBF (faithful axis-(0,3) sum) --
__global__ void __launch_bounds__(256) k_hat(const float* __restrict__ C,
                                             float* __restrict__ hat) {
    for (int p = threadIdx.x; p < 4096; p += 256) {
        const int i = p >> 6, j = p & 63;
        float s = 0.f;
        for (int b = 0; b < Bn; ++b) {
            const float* cb = C + (size_t)b * 128;
            const float dx = cb[i * 2] - cb[j * 2];
            const float dy = cb[i * 2 + 1] - cb[j * 2 + 1];
            s += dx * dx + dy * dy;
        }
        const float r = sqrtf((i == j) ? 1.0f : s);
        hat[p] = r * __logf(r);
    }
}

// ---------------- K5b: per-batch 67x67 solve (Gauss-Jordan, pivoting) ------
__global__ void __launch_bounds__(128) k_solve(const float* __restrict__ C,
                                               const float* __restrict__ hat,
                                               const float* __restrict__ Cp,
                                               float* __restrict__ T) {
    __shared__ float M[67 * 70];  // 69 used cols (67 + 2 RHS), stride 70
    __shared__ int pr;
    const int b = blockIdx.x, tid = threadIdx.x;
    for (int i = tid; i < 67 * 70; i += 128) M[i] = 0.f;
    __syncthreads();
    const float* cb = C + (size_t)b * 128;
    for (int idx = tid; idx < 4096; idx += 128) {
        const int i = idx >> 6, j = idx & 63;
        M[i * 70 + 3 + j] = hat[idx];
    }
    for (int i = tid; i < 64; i += 128) {
        M[i * 70 + 0] = 1.f;
        M[i * 70 + 1] = cb[i * 2];
        M[i * 70 + 2] = cb[i * 2 + 1];
        M[64 * 70 + 3 + i] = cb[i * 2];
        M[65 * 70 + 3 + i] = cb[i * 2 + 1];
        M[66 * 70 + 3 + i] = 1.f;
        M[i * 70 + 67] = Cp[(size_t)b * 128 + i * 2];
        M[i * 70 + 68] = Cp[(size_t)b * 128 + i * 2 + 1];
    }
    __syncthreads();
    for (int k = 0; k < 67; ++k) {
        if (tid == 0) {
            int best = k;
            float bv = fabsf(M[k * 70 + k]);
            for (int r = k + 1; r < 67; ++r) {
                const float v = fabsf(M[r * 70 + k]);
                if (v > bv) { bv = v; best = r; }
            }
            pr = best;
        }
        __syncthreads();
        const int best = pr;
        if (best != k)
            for (int c = tid; c < 69; c += 128) {
                const float t = M[k * 70 + c];
                M[k * 70 + c] = M[best * 70 + c];
                M[best * 70 + c] = t;
            }
        __syncthreads();
        const float piv = M[k * 70 + k];
        for (int r = tid; r < 67; r += 128) {
            if (r != k) {
                const float f = M[r * 70 + k] / piv;
                for (int c = k; c < 69; ++c) M[r * 70 + c] -= f * M[k * 70 + c];
            }
        }
        __syncthreads();
    }
    for (int k = tid; k < 67; k += 128) {
        const float d = M[k * 70 + k];
        T[((size_t)b * 67 + k) * 2] = M[k * 70 + 67] / d;
        T[((size_t)b * 67 + k) * 2 + 1] = M[k * 70 + 68] / d;
    }
}

// ---------------- K6: out = P_hat @ T --------------------------------------
__global__ void __launch_bounds__(256) k_final(const float* __restrict__ C,
                                               const float* __restrict__ T,
                                               float* __restrict__ out) {
    __shared__ float sC[128];
    __shared__ float sT[134];
    const int b = blockIdx.x, tid = threadIdx.x;
    if (tid < 128) sC[tid] = C[(size_t)b * 128 + tid];
    if (tid < 134) sT[tid] = T[(size_t)b * 134 + tid];
    __syncthreads();
    const int p = blockIdx.y * 256 + tid;
    if (p < NP) {
        const int i = p >> 5, j = p & 31;
        const float Px = (2.f * (float)i - 99.f) * 0.01f;     // (2i-99)/100
        const float Py = (2.f * (float)j - 31.f) * 0.03125f;  // (2j-31)/32
        float a0 = sT[0] + Px * sT[2] + Py * sT[4];
        float a1 = sT[1] + Px * sT[3] + Py * sT[5];
        for (int f = 0; f < 64; ++f) {
            const float dx = Px - sC[f * 2];
            const float dy = Py - sC[f * 2 + 1];
            const float d2 = fmaxf(dx * dx + dy * dy, 1e-20f);
            const float rbf = d2 * __logf(sqrtf(d2) + 1e-6f);
            a0 += rbf * sT[(3 + f) * 2];
            a1 += rbf * sT[(3 + f) * 2 + 1];
        }
        out[((size_t)b * NP + p) * 2] = a0;
        out[((size_t)b * NP + p) * 2 + 1] = a1;
    }
}

// ---------------- launch ---------------------------------------------------
extern "C" void kernel_launch(void* const* d_in, const int* in_sizes, int n_in,
                              void* d_out, int out_size, void* d_ws,
                              size_t ws_size, hipStream_t stream) {
    const float* bCp   = (const float*)d_in[0];
    const float* Cfeat = (const float*)d_in[1];
    const float* W_in  = (const float*)d_in[2];
    const float* b_in  = (const float*)d_in[3];
    const float* W_emb = (const float*)d_in[4];
    const float* b_emb = (const float*)d_in[5];
    const float* W_dn  = (const float*)d_in[6];
    const float* b_dn  = (const float*)d_in[7];
    const float* Wq = (const float*)d_in[8];
    const float* bq = (const float*)d_in[9];
    const float* Wk = (const float*)d_in[10];
    const float* bk = (const float*)d_in[11];
    const float* Wv = (const float*)d_in[12];
    const float* bv = (const float*)d_in[13];
    const float* Wo = (const float*)d_in[14];
    const float* bo = (const float*)d_in[15];
    const float* l1g = (const float*)d_in[16];
    const float* l1b = (const float*)d_in[17];
    const float* W1 = (const float*)d_in[18];
    const float* b1 = (const float*)d_in[19];
    const float* W2 = (const float*)d_in[20];
    const float* b2 = (const float*)d_in[21];
    const float* l2g = (const float*)d_in[22];
    const float* l2b = (const float*)d_in[23];

    float* ws = (float*)d_ws;
    const size_t KV = (size_t)Bn * Ln * 64;  // 16,777,216
    float* kp    = ws;
    float* vp    = kp + KV;
    float* qbuf  = vp + KV;          // 4096
    float* qpbuf = qbuf + 4096;      // 4096
    float* attnO = qpbuf + 4096;     // B*64*64 = 1,048,576
    float* Cbuf  = attnO + (size_t)Bn * 4096;  // B*64*2
    float* hat   = Cbuf + (size_t)Bn * 128;    // 4096
    float* Tbuf  = hat + 4096;                 // B*67*2

    k_query<<<1, 64, 0, stream>>>(W_emb, b_emb, Wq, bq, qbuf, qpbuf);
    k_kv<<<(Bn * Ln / 16) / 4, 128, 0, stream>>>(Cfeat, W_in, b_in, Wk, bk,
                                                 Wv, bv, kp, vp);
    k_attn<<<Bn * Hn, 128, 0, stream>>>(qpbuf, kp, vp, attnO);
    k_post<<<Bn, 128, 0, stream>>>(attnO, qbuf, Wo, bo, l1g, l1b, W1, b1, W2,
                                   b2, l2g, l2b, W_dn, b_dn, Cbuf);
    k_hat<<<1, 256, 0, stream>>>(Cbuf, hat);
    k_solve<<<Bn, 128, 0, stream>>>(Cbuf, hat, bCp, Tbuf);
    k_final<<<dim3(Bn, (NP + 255) / 256), 256, 0, stream>>>(Cbuf, Tbuf,
                                                            (float*)d_out);
}